// FGDTMloss_67130338836739
// MI455X (gfx1250) — compile-verified
//
#include <hip/hip_runtime.h>
#include <hip/hip_bf16.h>
#include <math.h>

// ---------------------------------------------------------------------------
// FGDTM loss for MI455X (gfx1250, wave32).
//  B=2 batches, C=2 channels, 128^3 volume, foreground channel only.
//  Pipeline:
//   K1: EDT pass along X (stride 2^14), fused with g = (y==1)?1e6:0.
//       Tiled 128x32 in LDS -> fully coalesced global access.
//   K2: EDT pass along Y (stride 2^7), same tiling.
//   K3: EDT pass along Z (contiguous lines) fused with sqrt + CE/Dice/dist
//       reductions. Wave-level f32-exact reduction via V_WMMA_F32_16X16X4_F32,
//       block combine in LDS, per-block partials to workspace (no atomics).
//   K4: final reduction of 32768x5 partials (WMMA wave-reduce) + scalar loss.
//  The EDT min-plus is tropical algebra -> VALU; WMMA is used where f32-exact
//  matrix math genuinely applies (sum reductions via B = ones).
// ---------------------------------------------------------------------------

typedef __attribute__((ext_vector_type(2))) float v2f;
typedef __attribute__((ext_vector_type(8))) float v8f;

#define VOL    (128 * 128 * 128)   // per batch, per channel
#define NB     2
#define LINE   128
#define TILEW  32                  // contiguous-z width of a cross-axis tile
#define NLINES ((NB * VOL) / LINE) // 32768
#define BIGINF 1000000.0f

// ---- wave32 full-precision cross-lane sum via V_WMMA_F32_16X16X4_F32 -------
// A(16x4): lane m (m<16) holds {part[m],0} (K=0,1); lane m+16 holds
// {part[m+16],0} (K=2,3). B = ones => D[m][n] = part[m] + part[m+16].
// Lane l's 8 accumulator VGPRs hold rows {0..7 | 8..15} of column (l&15), so
// per-lane VGPR sum + shfl_xor(16) = total of all 32 partials (in all lanes).
__device__ __forceinline__ float wave_sum_wmma(float part) {
    v2f a;    a.x = part; a.y = 0.0f;
    v2f ones; ones.x = 1.0f; ones.y = 1.0f;
    v8f c = {};
    c = __builtin_amdgcn_wmma_f32_16x16x4_f32(
            false, a, false, ones, (short)0, c, false, false);
    float s = c[0] + c[1] + c[2] + c[3] + c[4] + c[5] + c[6] + c[7];
    s += __shfl_xor(s, 16);
    return s;
}

// ---- shared compute for a tiled cross-axis pass ----------------------------
// tile[i*32+z] holds the 128x32 tile. Thread tid: z = tid&31, i0 = tid>>5;
// computes outputs i = i0 + 8r (r=0..15) of its z column and stores to g.
__device__ __forceinline__ void edt_tile_compute(const float* tile,
                                                 float* __restrict__ g,
                                                 size_t base, int s) {
    const int z  = threadIdx.x & 31;
    const int i0 = threadIdx.x >> 5;

    float best[16], d[16];
#pragma unroll
    for (int r = 0; r < 16; ++r) {
        best[r] = 3.402823466e38f;
        d[r]    = (float)(i0 + 8 * r);   // i_r - j  at j = 0
    }
    for (int j = 0; j < LINE; ++j) {
        float v = tile[(j << 5) + z];    // 32 consecutive words: bank-free
#pragma unroll
        for (int r = 0; r < 16; ++r) {
            float dr = d[r];
            best[r] = fminf(best[r], fmaf(dr, dr, v));
            d[r]    = dr - 1.0f;
        }
    }
#pragma unroll
    for (int r = 0; r < 16; ++r) {       // lanes -> 32 consecutive words
        g[base + ((size_t)(i0 + 8 * r) << s) + z] = best[r];
    }
}

__device__ __forceinline__ size_t tile_base(int t, int s) {
    const int outer = t >> (s - 5);                  // which 128*S slab
    const int c0    = (t & ((1 << (s - 5)) - 1)) << 5; // z-chunk offset
    return ((size_t)outer << (s + 7)) + (size_t)c0;
}

// ---- K1: X-axis pass (s=14) fused with mask init ---------------------------
__global__ void fgdtm_edt_x_init(const int* __restrict__ y,
                                 float* __restrict__ g) {
    __shared__ __align__(16) float tile[LINE * TILEW];
    const int s = 14;
    const size_t base = tile_base(blockIdx.x, s);
#pragma unroll
    for (int r = 0; r < 16; ++r) {
        int k = threadIdx.x + 256 * r;               // 0..4095
        int i = k >> 5, z = k & 31;
        tile[k] = (y[base + ((size_t)i << s) + z] == 1) ? BIGINF : 0.0f;
    }
    __syncthreads();
    edt_tile_compute(tile, g, base, s);
}

// ---- K2: Y-axis pass (s=7) -------------------------------------------------
__global__ void fgdtm_edt_y(float* __restrict__ g) {
    __shared__ __align__(16) float tile[LINE * TILEW];
    const int s = 7;
    const size_t base = tile_base(blockIdx.x, s);
#pragma unroll
    for (int r = 0; r < 16; ++r) {
        int k = threadIdx.x + 256 * r;
        int i = k >> 5, z = k & 31;
        tile[k] = g[base + ((size_t)i << s) + z];
    }
    __syncthreads();
    edt_tile_compute(tile, g, base, s);
}

// ---- K3: Z-axis pass fused with sqrt + loss-term reductions ----------------
// One contiguous 128-line per 128-thread block (4 waves). EDT result never
// touches memory; per-block partial sums written to ws (no atomics).
__global__ void fgdtm_edt_z_loss(const float* __restrict__ g,
                                 const float* __restrict__ outputs,
                                 const float* __restrict__ outputs_dist,
                                 const int*   __restrict__ y,
                                 float* __restrict__ part) {
    __shared__ __align__(16) float line[LINE];
    __shared__ float wsum[4][5];

    const int l = blockIdx.x;            // line id, 32768 lines
    const int i = threadIdx.x;           // 0..127
    const size_t idx = (size_t)l * LINE + i;

    line[i] = g[idx];
    __syncthreads();

    // brute-force lower envelope along z (float4 LDS broadcasts)
    const float fi = (float)i;
    float best = 3.402823466e38f;
    const float4* line4 = (const float4*)line;
#pragma unroll 8
    for (int j4 = 0; j4 < LINE / 4; ++j4) {
        float4 v = line4[j4];
        float d0 = fi - (float)(4 * j4);
        float d1 = d0 - 1.0f;
        float d2 = d0 - 2.0f;
        float d3 = d0 - 3.0f;
        best = fminf(best, fmaf(d0, d0, v.x));
        best = fminf(best, fmaf(d1, d1, v.y));
        best = fminf(best, fmaf(d2, d2, v.z));
        best = fminf(best, fmaf(d3, d3, v.w));
    }
    const float gt = sqrtf(best);

    // loss terms for this voxel (branchless -> EXEC all-ones at the WMMA)
    const int b = l >> 14;               // batch = idx / VOL
    const size_t v = idx - (size_t)b * VOL;
    const float xv  = outputs     [(size_t)b * 2 * VOL + VOL + v];
    const float odv = outputs_dist[(size_t)b * 2 * VOL + VOL + v];
    const float tv  = (y[idx] == 1) ? 1.0f : 0.0f;

    const float ax = fabsf(xv);
    const float ce = fmaxf(xv, 0.0f) - xv * tv + log1pf(expf(-ax));
    const float p  = 1.0f / (1.0f + expf(-xv));

    float q[5];
    q[0] = ce;
    q[1] = p * tv;
    q[2] = p;
    q[3] = tv;
    q[4] = fabsf(odv - gt) * tv;

    const int wave = threadIdx.x >> 5;
    const int lane = threadIdx.x & 31;
#pragma unroll
    for (int k = 0; k < 5; ++k) {
        float t = wave_sum_wmma(q[k]);   // all lanes hold the wave total
        if (lane == 0) wsum[wave][k] = t;
    }
    __syncthreads();
    if (threadIdx.x < 5) {
        part[threadIdx.x * NLINES + l] =
            wsum[0][threadIdx.x] + wsum[1][threadIdx.x] +
            wsum[2][threadIdx.x] + wsum[3][threadIdx.x];
    }
}

// ---- K4: fold 5 x 32768 partials into the scalar loss ----------------------
// Partials: part[q*32768 + l]; lines [0,16384) = batch 0, [16384,32768) = b1.
__global__ void fgdtm_final(const float* __restrict__ part,
                            const float* __restrict__ w,
                            float* __restrict__ out) {
    __shared__ float red[8];
    __shared__ float totals[10];

    const int wave = threadIdx.x >> 5;
    const int lane = threadIdx.x & 31;

    for (int qb = 0; qb < 10; ++qb) {
        const int q = qb >> 1, b = qb & 1;
        const float* p = part + q * NLINES + b * (NLINES / 2);
        float s = 0.0f;
        for (int i = threadIdx.x; i < NLINES / 2; i += 256) s += p[i];
        s = wave_sum_wmma(s);
        if (lane == 0) red[wave] = s;
        __syncthreads();
        if (threadIdx.x == 0) {
            totals[qb] = red[0] + red[1] + red[2] + red[3] +
                         red[4] + red[5] + red[6] + red[7];
        }
        __syncthreads();
    }

    if (threadIdx.x == 0) {
        // qb layout: ce:0,1  pt:2,3  p:4,5  t:6,7  d:8,9
        float loss_ce = (totals[0] + totals[1]) / (float)((size_t)NB * VOL);

        float dice = 0.0f;
#pragma unroll
        for (int b = 0; b < NB; ++b) {
            dice += (2.0f * totals[2 + b] + 1.0f) /
                    (totals[4 + b] + totals[6 + b] + 1.0f);
        }
        float loss_dice = 1.0f - dice / (float)NB;

        float msum = totals[6] + totals[7];
        float dsum = totals[8] + totals[9];
        float loss_dist = (msum == 0.0f) ? 0.0f : dsum / fmaxf(msum, 1e-12f);

        out[0] = loss_ce + loss_dice + w[0] * loss_dist;
    }
}

// ---------------------------------------------------------------------------
extern "C" void kernel_launch(void* const* d_in, const int* in_sizes, int n_in,
                              void* d_out, int out_size, void* d_ws, size_t ws_size,
                              hipStream_t stream) {
    (void)in_sizes; (void)n_in; (void)out_size; (void)ws_size;

    const float* outputs      = (const float*)d_in[0];  // (2,2,128^3) f32
    const float* outputs_dist = (const float*)d_in[1];  // (2,2,128^3) f32
    const int*   y            = (const int*)  d_in[2];  // (2,1,128^3) i32
    const float* wgt          = (const float*)d_in[3];  // scalar

    float* part = (float*)d_ws;                           // 5*32768 f32 (640KB)
    float* g    = (float*)((char*)d_ws + (1 << 20));      // NB*VOL f32 (16.8MB)

    const int ntiles = (NB * VOL) / (LINE * TILEW);       // 1024 tiles per pass
    fgdtm_edt_x_init<<<ntiles, 256, 0, stream>>>(y, g);   // X axis + mask init
    fgdtm_edt_y     <<<ntiles, 256, 0, stream>>>(g);      // Y axis

    fgdtm_edt_z_loss<<<NLINES, LINE, 0, stream>>>(g, outputs, outputs_dist, y,
                                                  part);  // Z axis + loss terms

    fgdtm_final<<<1, 256, 0, stream>>>(part, wgt, (float*)d_out);
}